// PrivacyGNNWithExplanation_27212912787887
// MI455X (gfx1250) — compile-verified
//
#include <hip/hip_runtime.h>
#include <math.h>

// ---------------- CDNA5 fp32 WMMA (V_WMMA_F32_16X16X4_F32) helpers ----------
typedef __attribute__((ext_vector_type(2))) float v2f;
typedef __attribute__((ext_vector_type(8))) float v8f;

// A fragment (ISA 7.12.2, 32-bit A 16x4): lanes 0-15 hold K={k0,k0+1},
// lanes 16-31 hold K={k0+2,k0+3}; M = lane & 15.  Contiguous pair -> b64 load.
__device__ __forceinline__ v2f a_frag(const float* __restrict__ A, int lda, int k0, int lane) {
  int m = lane & 15, kh = (lane >> 4) << 1;
  const float* p = A + m * lda + k0 + kh;
  v2f a; a.x = p[0]; a.y = p[1];
  return a;
}

// Stage row-major B[K x NTOT] into LDS, packed so each lane's fragment
// (two K rows, ldb apart in global) becomes ONE contiguous 8-byte ds load:
//   sB[((k>>1)*NTOT + n)*2 + (k&1)] = B[k*NTOT + n]
template <int K, int NTOT>
__device__ __forceinline__ void stage_B(const float* __restrict__ B, float* sB) {
  for (int idx = threadIdx.x; idx < K * NTOT; idx += blockDim.x) {
    int k = idx / NTOT, n = idx % NTOT;
    sB[(((k >> 1) * NTOT) + n) * 2 + (k & 1)] = B[idx];
  }
  __syncthreads();
}

// B fragment from packed LDS: single ds_load_b64 per fragment.
__device__ __forceinline__ v2f b_frag_lds(const float* sB, int NTOT, int k0, int n0, int lane) {
  int n = lane & 15, kp = (k0 >> 1) + (lane >> 4);
  const float* p = sB + ((size_t)kp * NTOT + n0 + n) * 2;
  v2f b; b.x = p[0]; b.y = p[1];
  return b;
}

// MT-tiled wave GEMM, B served from packed LDS. MT=2/NT=4 -> 64 accum VGPRs,
// per 4-K step: 2 global b64 (A) + 4 ds b64 (B) for 8 WMMAs.
template <int KTOT, int NT, int MT>
__device__ __forceinline__ void wave_gemm_ldsB(const float* __restrict__ A, int lda,
                                               const int* arow,
                                               const float* sB, int NTOT, int n0base,
                                               v8f (&acc)[MT][NT], int lane) {
#pragma unroll 4
  for (int k0 = 0; k0 < KTOT; k0 += 4) {
    v2f a[MT];
#pragma unroll
    for (int mt = 0; mt < MT; ++mt)
      a[mt] = a_frag(A + (size_t)arow[mt] * lda, lda, k0, lane);
#pragma unroll
    for (int t = 0; t < NT; ++t) {
      v2f b = b_frag_lds(sB, NTOT, k0, n0base + t * 16, lane);
#pragma unroll
      for (int mt = 0; mt < MT; ++mt)
        acc[mt][t] = __builtin_amdgcn_wmma_f32_16x16x4_f32(false, a[mt], false, b, (short)0,
                                                           acc[mt][t], false, false);
    }
  }
}

template <int NT>
__device__ __forceinline__ void init_bias(v8f* acc, const float* __restrict__ b, int lane) {
  int n = lane & 15;
#pragma unroll
  for (int t = 0; t < NT; ++t) {
    float bv = b[t * 16 + n];
    v8f v;
#pragma unroll
    for (int r = 0; r < 8; ++r) v[r] = bv;
    acc[t] = v;
  }
}

// C/D layout: VGPR r holds row M = r + 8*(lane>=16), column N = lane & 15.
template <int NT, bool RELU>
__device__ __forceinline__ void store_tile(float* __restrict__ O, int ldo,
                                           const v8f* acc, int lane) {
  int n = lane & 15, mh = lane >> 4;
#pragma unroll
  for (int t = 0; t < NT; ++t)
#pragma unroll
    for (int r = 0; r < 8; ++r) {
      float v = acc[t][r];
      if (RELU) v = fmaxf(v, 0.0f);
      O[(r + 8 * mh) * ldo + t * 16 + n] = v;
    }
}

__device__ __forceinline__ void atomicMaxF(float* addr, float v) {
  if (v >= 0.0f) atomicMax((int*)addr, __float_as_int(v));
  else           atomicMin((unsigned int*)addr, (unsigned int)__float_as_int(v));
}

__device__ __forceinline__ void make_rows2(int row0, int nrows, int* arow, bool* act) {
#pragma unroll
  for (int mt = 0; mt < 2; ++mt) {
    int r = row0 + mt * 16;
    act[mt] = r < nrows;                    // wave-uniform
    arow[mt] = act[mt] ? r : (nrows - 16);  // clamp tail to valid rows
  }
}

// ---------------- dense kernels (MT=2, 32 rows/wave, 256 rows/block) --------
// t1 = relu(x @ fa_W1 + b1)   [N,128]x[128,64]
__global__ void k_fa1(const float* __restrict__ x, const float* __restrict__ W1,
                      const float* __restrict__ b1, float* __restrict__ t1, int nrows) {
  __shared__ float sB[128 * 64];
  stage_B<128, 64>(W1, sB);
  int wave = (blockIdx.x * blockDim.x + threadIdx.x) >> 5;
  int lane = threadIdx.x & 31;
  int row0 = wave * 32;
  if (row0 < nrows) {
    int arow[2]; bool act[2];
    make_rows2(row0, nrows, arow, act);
    v8f acc[2][4];
#pragma unroll
    for (int mt = 0; mt < 2; ++mt) init_bias<4>(acc[mt], b1, lane);
    wave_gemm_ldsB<128, 4, 2>(x, 128, arow, sB, 64, 0, acc, lane);
#pragma unroll
    for (int mt = 0; mt < 2; ++mt)
      if (act[mt]) store_tile<4, true>(t1 + (size_t)arow[mt] * 64, 64, acc[mt], lane);
  }
}

// xf = x * sigmoid(t1 @ fa_W2 + b2)   [N,64]x[64,128]; two 64-col halves
__global__ void k_fa2(const float* __restrict__ x, const float* __restrict__ t1,
                      const float* __restrict__ W2, const float* __restrict__ b2,
                      float* __restrict__ xf, int nrows) {
  __shared__ float sB[64 * 128];
  stage_B<64, 128>(W2, sB);
  int wave = (blockIdx.x * blockDim.x + threadIdx.x) >> 5;
  int lane = threadIdx.x & 31;
  int row0 = wave * 32;
  if (row0 < nrows) {
    int arow[2]; bool act[2];
    make_rows2(row0, nrows, arow, act);
    int n = lane & 15, mh = lane >> 4;
    for (int nh = 0; nh < 2; ++nh) {          // column halves, 64 accum VGPRs live
      v8f acc[2][4];
#pragma unroll
      for (int mt = 0; mt < 2; ++mt) init_bias<4>(acc[mt], b2 + nh * 64, lane);
      wave_gemm_ldsB<64, 4, 2>(t1, 64, arow, sB, 128, nh * 64, acc, lane);
#pragma unroll
      for (int mt = 0; mt < 2; ++mt) {
        if (!act[mt]) continue;
#pragma unroll
        for (int t = 0; t < 4; ++t)
#pragma unroll
          for (int r = 0; r < 8; ++r) {
            int row = arow[mt] + r + 8 * mh, col = nh * 64 + t * 16 + n;
            float s = 1.0f / (1.0f + expf(-acc[mt][t][r]));
            xf[(size_t)row * 128 + col] = x[(size_t)row * 128 + col] * s;
          }
      }
    }
  }
}

// h = xf @ proj_W + proj_b   [N,128]x[128,64]
__global__ void k_proj(const float* __restrict__ xf, const float* __restrict__ W,
                       const float* __restrict__ b, float* __restrict__ h, int nrows) {
  __shared__ float sB[128 * 64];
  stage_B<128, 64>(W, sB);
  int wave = (blockIdx.x * blockDim.x + threadIdx.x) >> 5;
  int lane = threadIdx.x & 31;
  int row0 = wave * 32;
  if (row0 < nrows) {
    int arow[2]; bool act[2];
    make_rows2(row0, nrows, arow, act);
    v8f acc[2][4];
#pragma unroll
    for (int mt = 0; mt < 2; ++mt) init_bias<4>(acc[mt], b, lane);
    wave_gemm_ldsB<128, 4, 2>(xf, 128, arow, sB, 64, 0, acc, lane);
#pragma unroll
    for (int mt = 0; mt < 2; ++mt)
      if (act[mt]) store_tile<4, false>(h + (size_t)arow[mt] * 64, 64, acc[mt], lane);
  }
}

// z = h @ gat_W[l]   [N,64]x[64,64]   (bias applied post-aggregation)
__global__ void k_z(const float* __restrict__ h, const float* __restrict__ W,
                    float* __restrict__ z, int nrows) {
  __shared__ float sB[64 * 64];
  stage_B<64, 64>(W, sB);
  int wave = (blockIdx.x * blockDim.x + threadIdx.x) >> 5;
  int lane = threadIdx.x & 31;
  int row0 = wave * 32;
  if (row0 < nrows) {
    int arow[2]; bool act[2];
    make_rows2(row0, nrows, arow, act);
    v8f acc[2][4];
#pragma unroll
    for (int mt = 0; mt < 2; ++mt)
#pragma unroll
      for (int t = 0; t < 4; ++t) {
        v8f v;
#pragma unroll
        for (int r = 0; r < 8; ++r) v[r] = 0.f;
        acc[mt][t] = v;
      }
    wave_gemm_ldsB<64, 4, 2>(h, 64, arow, sB, 64, 0, acc, lane);
#pragma unroll
    for (int mt = 0; mt < 2; ++mt)
      if (act[mt]) store_tile<4, false>(z + (size_t)arow[mt] * 64, 64, acc[mt], lane);
  }
}

// as/ad = z @ a_src / z @ a_dst
__global__ void k_scores(const float* __restrict__ z, const float* __restrict__ av,
                         const float* __restrict__ aw, float* __restrict__ as_,
                         float* __restrict__ ad_, int nN) {
  int i = blockIdx.x * blockDim.x + threadIdx.x;
  if (i >= nN) return;
  const float* zr = z + (size_t)i * 64;
  float sa = 0.f, sd = 0.f;
#pragma unroll
  for (int c = 0; c < 64; ++c) { sa += zr[c] * av[c]; sd += zr[c] * aw[c]; }
  as_[i] = sa; ad_[i] = sd;
}

// ---------------- edge-phase kernels ----------------------------------------
__global__ void k_layer_init(float* __restrict__ hnew, float* __restrict__ m,
                             float* __restrict__ den, int nN) {
  int i = blockIdx.x * blockDim.x + threadIdx.x;
  if (i < nN * 64) hnew[i] = 0.0f;
  if (i < nN) { m[i] = -__builtin_huge_valf(); den[i] = 0.0f; }
}

__device__ __forceinline__ void edge_sd(const long long* __restrict__ ei, int i, int nE,
                                        int& s, int& d) {
  if (i < nE) { s = (int)ei[i]; d = (int)ei[nE + i]; }
  else        { s = d = i - nE; }  // self loops
}

__global__ void k_edge_max(const long long* __restrict__ ei, const float* __restrict__ as_,
                           const float* __restrict__ ad_, float* __restrict__ m,
                           int nE, int nN) {
  int i = blockIdx.x * blockDim.x + threadIdx.x;
  if (i >= nE + nN) return;
  int s, d; edge_sd(ei, i, nE, s, d);
  float e = as_[s] + ad_[d];
  e = (e > 0.f) ? e : 0.2f * e;   // leaky_relu 0.2
  atomicMaxF(&m[d], e);
}

__global__ void k_clamp_m(float* __restrict__ m, int nN) {
  int i = blockIdx.x * blockDim.x + threadIdx.x;
  if (i >= nN) return;
  float v = m[i];
  m[i] = (fabsf(v) <= 3.402823e38f) ? v : 0.0f;  // where(isfinite, m, 0)
}

__global__ void k_edge_sum(const long long* __restrict__ ei, const float* __restrict__ as_,
                           const float* __restrict__ ad_, const float* __restrict__ m,
                           float* __restrict__ den, int nE, int nN) {
  int i = blockIdx.x * blockDim.x + threadIdx.x;
  if (i >= nE + nN) return;
  int s, d; edge_sd(ei, i, nE, s, d);
  float e = as_[s] + ad_[d];
  e = (e > 0.f) ? e : 0.2f * e;
  atomicAdd(&den[d], expf(e - m[d]));
}

// one wave per edge; lane handles columns {lane, lane+32}; z/hnew stay in L2
__global__ void k_edge_scatter(const long long* __restrict__ ei, const float* __restrict__ as_,
                               const float* __restrict__ ad_, const float* __restrict__ m,
                               const float* __restrict__ den, const float* __restrict__ z,
                               float* __restrict__ hnew, int nE, int nN) {
  int g = blockIdx.x * blockDim.x + threadIdx.x;
  int edge = g >> 5, lane = g & 31;
  if (edge >= nE + nN) return;
  int s, d; edge_sd(ei, edge, nE, s, d);
  float e = as_[s] + ad_[d];
  e = (e > 0.f) ? e : 0.2f * e;
  float alpha = expf(e - m[d]) / den[d];
  const float* zr = z + (size_t)s * 64;
  float* hr = hnew + (size_t)d * 64;
  atomicAdd(&hr[lane],      zr[lane]      * alpha);
  atomicAdd(&hr[lane + 32], zr[lane + 32] * alpha);
}

__global__ void k_bn_relu(const float* __restrict__ hnew, const float* __restrict__ gb,
                          const float* __restrict__ gamma, const float* __restrict__ beta,
                          const float* __restrict__ mean, const float* __restrict__ var,
                          float* __restrict__ h, int nN) {
  int i = blockIdx.x * blockDim.x + threadIdx.x;
  if (i >= nN * 64) return;
  int c = i & 63;
  float v = hnew[i] + gb[c];
  v = (v - mean[c]) * rsqrtf(var[c] + 1e-5f) * gamma[c] + beta[c];
  h[i] = fmaxf(v, 0.0f);
}

// ---------------- structural features ---------------------------------------
__global__ void k_struct_init(float* __restrict__ deg, float* __restrict__ isum,
                              float* __restrict__ scal, int nN) {
  int i = blockIdx.x * blockDim.x + threadIdx.x;
  if (i < nN) { deg[i] = 0.0f; isum[i] = 0.0f; }
  if (i == 0) { scal[0] = 1.0f; scal[1] = 1e-12f; }  // max(deg.max,1), max(infl.max,1e-12)
}
__global__ void k_deg(const long long* __restrict__ ei, float* __restrict__ deg, int nE) {
  int i = blockIdx.x * blockDim.x + threadIdx.x;
  if (i >= nE) return;
  atomicAdd(&deg[(int)ei[i]], 1.0f);
}
__global__ void k_degmax(const float* __restrict__ deg, float* __restrict__ scal, int nN) {
  int i = blockIdx.x * blockDim.x + threadIdx.x;
  if (i >= nN) return;
  atomicMaxF(&scal[0], deg[i]);
}
__global__ void k_isum(const long long* __restrict__ ei, const float* __restrict__ deg,
                       float* __restrict__ isum, int nE) {
  int i = blockIdx.x * blockDim.x + threadIdx.x;
  if (i >= nE) return;
  atomicAdd(&isum[(int)ei[i]], deg[(int)ei[nE + i]]);
}
__global__ void k_infl(const float* __restrict__ deg, const float* __restrict__ isum,
                       float* __restrict__ infl, float* __restrict__ scal, int nN) {
  int i = blockIdx.x * blockDim.x + threadIdx.x;
  if (i >= nN) return;
  float d = deg[i];
  float ir = (d > 0.f) ? isum[i] / fmaxf(d, 1.0f) : 0.0f;
  infl[i] = ir;
  atomicMaxF(&scal[1], ir);
}
// se = relu(sf @ se_W1 + b1) @ se_W2 + b2 ; sf = [deg/degmax, 0, infl/inflmax]
__global__ void k_se(const float* __restrict__ deg, const float* __restrict__ infl,
                     const float* __restrict__ scal, const float* __restrict__ W1,
                     const float* __restrict__ b1, const float* __restrict__ W2,
                     const float* __restrict__ b2, float* __restrict__ se, int nN) {
  int i = blockIdx.x * blockDim.x + threadIdx.x;
  if (i >= nN) return;
  float f0 = deg[i] / scal[0];
  float f2 = infl[i] / scal[1];
  float hb[32];
#pragma unroll
  for (int j = 0; j < 32; ++j)
    hb[j] = fmaxf(f0 * W1[j] + f2 * W1[64 + j] + b1[j], 0.0f);  // middle row * 0
  float* o = se + (size_t)i * 64;
  for (int c = 0; c < 64; ++c) {
    float s = b2[c];
#pragma unroll
    for (int j = 0; j < 32; ++j) s += hb[j] * W2[j * 64 + c];
    o[c] = s;
  }
}

// ---------------- fused output MLP (two WMMA GEMMs + sigmoid) ---------------
__global__ void k_out(const float* __restrict__ h, const float* __restrict__ se,
                      const float* __restrict__ W1, const float* __restrict__ b1,
                      const float* __restrict__ W2, const float* __restrict__ b2,
                      const float* __restrict__ W3, const float* __restrict__ b3,
                      float* __restrict__ out, int nrows) {
  __shared__ float sB1[128 * 64];
  __shared__ float sB2[64 * 32];
  __shared__ float s1[8][16 * 64];
  __shared__ float s2[8][16 * 32];
  stage_B<128, 64>(W1, sB1);        // includes __syncthreads
  for (int idx = threadIdx.x; idx < 64 * 32; idx += blockDim.x) {
    int k = idx / 32, n = idx % 32;
    sB2[(((k >> 1) * 32) + n) * 2 + (k & 1)] = W2[idx];
  }
  __syncthreads();

  int w = threadIdx.x >> 5, lane = threadIdx.x & 31;
  int wave = blockIdx.x * 8 + w;
  int row0 = wave * 16;
  bool active = row0 < nrows;       // wave-uniform; EXEC all-ones inside branches
  int n = lane & 15, mh = lane >> 4;
  if (active) {
    // o1 = relu([h | se] @ W1 + b1), K=128 split across the two source buffers
    v8f acc[4];
    init_bias<4>(acc, b1, lane);
#pragma unroll 4
    for (int k0 = 0; k0 < 128; k0 += 4) {
      const float* A = (k0 < 64) ? (h + (size_t)row0 * 64 + k0)
                                 : (se + (size_t)row0 * 64 + (k0 - 64));
      int kh = (lane >> 4) << 1;
      const float* p = A + (lane & 15) * 64 + kh;
      v2f a; a.x = p[0]; a.y = p[1];
#pragma unroll
      for (int t = 0; t < 4; ++t) {
        v2f b = b_frag_lds(sB1, 64, k0, t * 16, lane);
        acc[t] = __builtin_amdgcn_wmma_f32_16x16x4_f32(false, a, false, b, (short)0,
                                                       acc[t], false, false);
      }
    }
#pragma unroll
    for (int t = 0; t < 4; ++t)
#pragma unroll
      for (int r = 0; r < 8; ++r)
        s1[w][(r + 8 * mh) * 64 + t * 16 + n] = fmaxf(acc[t][r], 0.0f);
  }
  __syncthreads();
  if (active) {
    v8f acc2[2];
    init_bias<2>(acc2, b2, lane);
#pragma unroll 4
    for (int k0 = 0; k0 < 64; k0 += 4) {
      v2f a = a_frag(&s1[w][0], 64, k0, lane);
#pragma unroll
      for (int t = 0; t < 2; ++t) {
        v2f b = b_frag_lds(sB2, 32, k0, t * 16, lane);
        acc2[t] = __builtin_amdgcn_wmma_f32_16x16x4_f32(false, a, false, b, (short)0,
                                                        acc2[t], false, false);
      }
    }
#pragma unroll
    for (int t = 0; t < 2; ++t)
#pragma unroll
      for (int r = 0; r < 8; ++r)
        s2[w][(r + 8 * mh) * 32 + t * 16 + n] = fmaxf(acc2[t][r], 0.0f);
  }
  __syncthreads();
  if (active && lane < 16) {
    float s = b3[0];
#pragma unroll
    for (int j = 0; j < 32; ++j) s += s2[w][lane * 32 + j] * W3[j];
    out[row0 + lane] = 1.0f / (1.0f + expf(-s));
  }
}

// ---------------- launcher ---------------------------------------------------
extern "C" void kernel_launch(void* const* d_in, const int* in_sizes, int n_in,
                              void* d_out, int out_size, void* d_ws, size_t ws_size,
                              hipStream_t stream) {
  const float*     x     = (const float*)d_in[0];
  const long long* ei    = (const long long*)d_in[1];
  const float* fa_W1     = (const float*)d_in[2];
  const float* fa_b1     = (const float*)d_in[3];
  const float* fa_W2     = (const float*)d_in[4];
  const float* fa_b2     = (const float*)d_in[5];
  const float* proj_W    = (const float*)d_in[6];
  const float* proj_b    = (const float*)d_in[7];
  const float* gat_W     = (const float*)d_in[8];
  const float* gat_asrc  = (const float*)d_in[9];
  const float* gat_adst  = (const float*)d_in[10];
  const float* gat_b     = (const float*)d_in[11];
  const float* bn_gamma  = (const float*)d_in[12];
  const float* bn_beta   = (const float*)d_in[13];
  const float* bn_mean   = (const float*)d_in[14];
  const float* bn_var    = (const float*)d_in[15];
  const float* se_W1     = (const float*)d_in[16];
  const float* se_b1     = (const float*)d_in[17];
  const float* se_W2     = (const float*)d_in[18];
  const float* se_b2     = (const float*)d_in[19];
  const float* out_W1    = (const float*)d_in[20];
  const float* out_b1    = (const float*)d_in[21];
  const float* out_W2    = (const float*)d_in[22];
  const float* out_b2    = (const float*)d_in[23];
  const float* out_W3    = (const float*)d_in[24];
  const float* out_b3    = (const float*)d_in[25];
  float* out = (float*)d_out;

  const int N = in_sizes[0] / 128;
  const int E = in_sizes[1] / 2;
  const int Etot = E + N;

  // workspace bump allocation (floats); xf region is reused for hnew|se later
  float* ws = (float*)d_ws;
  size_t o = 0;
  float* xf   = ws + o; o += (size_t)N * 128;  // xf, later [hnew | se]
  float* h    = ws + o; o += (size_t)N * 64;
  float* zz   = ws + o; o += (size_t)N * 64;   // t1, later z
  float* as_  = ws + o; o += N;
  float* ad_  = ws + o; o += N;
  float* mm   = ws + o; o += N;
  float* den  = ws + o; o += N;
  float* deg  = ws + o; o += N;
  float* isum = ws + o; o += N;
  float* infl = ws + o; o += N;
  float* scal = ws + o; o += 2;
  float* hnew = xf;
  float* se   = xf + (size_t)N * 64;

  const int B = 256;
  int wb32   = ((((N + 31) / 32) * 32) + B - 1) / B;   // MT=2 kernels (32 rows/wave)
  int nodeN  = (N + B - 1) / B;
  int node64 = (N * 64 + B - 1) / B;
  int edgeB  = (Etot + B - 1) / B;
  int edgeE  = (E + B - 1) / B;
  int scatB  = (int)(((size_t)Etot * 32 + B - 1) / B);

  // feature attention + projection
  k_fa1 <<<wb32, B, 0, stream>>>(x, fa_W1, fa_b1, zz, N);
  k_fa2 <<<wb32, B, 0, stream>>>(x, zz, fa_W2, fa_b2, xf, N);
  k_proj<<<wb32, B, 0, stream>>>(xf, proj_W, proj_b, h, N);

  // 3 GAT layers
  for (int l = 0; l < 3; ++l) {
    k_layer_init  <<<node64, B, 0, stream>>>(hnew, mm, den, N);
    k_z           <<<wb32, B, 0, stream>>>(h, gat_W + (size_t)l * 64 * 64, zz, N);
    k_scores      <<<nodeN, B, 0, stream>>>(zz, gat_asrc + l * 64, gat_adst + l * 64, as_, ad_, N);
    k_edge_max    <<<edgeB, B, 0, stream>>>(ei, as_, ad_, mm, E, N);
    k_clamp_m     <<<nodeN, B, 0, stream>>>(mm, N);
    k_edge_sum    <<<edgeB, B, 0, stream>>>(ei, as_, ad_, mm, den, E, N);
    k_edge_scatter<<<scatB, B, 0, stream>>>(ei, as_, ad_, mm, den, zz, hnew, E, N);
    k_bn_relu     <<<node64, B, 0, stream>>>(hnew, gat_b + l * 64, bn_gamma + l * 64,
                                             bn_beta + l * 64, bn_mean + l * 64,
                                             bn_var + l * 64, h, N);
  }

  // structural features + SE MLP
  k_struct_init<<<nodeN, B, 0, stream>>>(deg, isum, scal, N);
  k_deg        <<<edgeE, B, 0, stream>>>(ei, deg, E);
  k_degmax     <<<nodeN, B, 0, stream>>>(deg, scal, N);
  k_isum       <<<edgeE, B, 0, stream>>>(ei, deg, isum, E);
  k_infl       <<<nodeN, B, 0, stream>>>(deg, isum, infl, scal, N);
  k_se         <<<nodeN, B, 0, stream>>>(deg, infl, scal, se_W1, se_b1, se_W2, se_b2, se, N);

  // fused output MLP
  int out_blocks = ((N + 15) / 16 + 7) / 8;
  k_out<<<out_blocks, B, 0, stream>>>(h, se, out_W1, out_b1, out_W2, out_b2,
                                      out_W3, out_b3, out, N);
}